// LightGCN_implicit_4355096838837
// MI455X (gfx1250) — compile-verified
//
#include <hip/hip_runtime.h>
#include <hip/hip_bf16.h>

#define U_CNT 50000
#define I_CNT 25000
#define DIM   128
#define N_CNT (U_CNT + I_CNT)
#define BATCH 4096
#define EDGES_PER_WAVE 4

typedef __attribute__((ext_vector_type(2))) float v2f;
typedef __attribute__((ext_vector_type(8))) float v8f;

// ---------------------------------------------------------------- zero h ----
__global__ void lgcn_zero_f4(float4* __restrict__ p, int n4) {
    int i = blockIdx.x * blockDim.x + threadIdx.x;
    if (i < n4) p[i] = make_float4(0.f, 0.f, 0.f, 0.f);
}

// ------------------------------------------------------- SpMM scatter-add ----
// One wave32 per EDGES_PER_WAVE edges: lanes split the 128-float source row
// into float4 chunks, scale by the edge weight, and scatter-add into h[row]
// (h is 38.4 MB -> resident in the 192 MB L2; atomics resolve in L2).
__global__ void lgcn_spmm_edges(const int*   __restrict__ adj_row,
                                const int*   __restrict__ adj_col,
                                const float* __restrict__ adj_vals,
                                const float* __restrict__ user_emb,
                                const float* __restrict__ item_emb,
                                float*       __restrict__ h,
                                int nE) {
    const int gid  = blockIdx.x * blockDim.x + threadIdx.x;
    const int wave = gid >> 5;          // uniform across the wave
    const int lane = gid & 31;
    const int e0   = __builtin_amdgcn_readfirstlane(wave) * EDGES_PER_WAVE;
    if (e0 >= nE) return;               // wave-uniform guard

    // Stream-ahead prefetch of the edge arrays (global_prefetch_b8).
    __builtin_prefetch(&adj_row[e0 + 4096], 0, 1);
    __builtin_prefetch(&adj_col[e0 + 4096], 0, 1);
    __builtin_prefetch(&adj_vals[e0 + 4096], 0, 1);

#pragma unroll
    for (int i = 0; i < EDGES_PER_WAVE; ++i) {
        const int eu = e0 + i;          // uniform (scalar-load path)
        if (eu >= nE) break;            // uniform guard (tail wave only)

        const int   r = adj_row[eu];
        const int   c = adj_col[eu];
        const float v = adj_vals[eu];

        const float* xrow = (c < U_CNT) ? (user_emb + (size_t)c * DIM)
                                        : (item_emb + (size_t)(c - U_CNT) * DIM);
        const float4 xv = *(const float4*)(xrow + lane * 4);

        float* hrow = h + (size_t)r * DIM + lane * 4;
        unsafeAtomicAdd(hrow + 0, v * xv.x);
        unsafeAtomicAdd(hrow + 1, v * xv.y);
        unsafeAtomicAdd(hrow + 2, v * xv.z);
        unsafeAtomicAdd(hrow + 3, v * xv.w);
    }
}

// ------------------------------------------- i_final blend via f32 WMMA ----
// final = 0.25*item_emb + 0.75*h[U:].  Per 16x16 tile: 4 chained
// V_WMMA_F32_16X16X4_F32 with A = h-tile column chunks and B = 0.75*I blocks.
// Each output element gets exactly one nonzero product, so this is exact.
template <bool FULL>
__device__ __forceinline__ void lgcn_item_tile(const float* __restrict__ item_emb,
                                               const float* __restrict__ h,
                                               float*       __restrict__ i_final,
                                               int row0, int col0, int lane) {
    const int n    = lane & 15;         // C/D layout: lane -> column
    const int half = lane >> 4;         // 0: rows 0..7, 1: rows 8..15

    // C = 0.25 * ego tile (C/D layout: VGPR v holds row v + 8*half, col n).
    const int mbase = row0 + 8 * half;
    v8f acc;
#pragma unroll
    for (int v = 0; v < 8; ++v) {
        int m  = mbase + v;
        int rm = FULL ? m : (m < I_CNT ? m : I_CNT - 1);
        acc[v] = 0.25f * item_emb[(size_t)rm * DIM + col0 + n];
    }

    // A layout (16x4 f32): lane -> M = n ; VGPR0 -> K = 2*half, VGPR1 -> K+1.
    int ra = row0 + n;
    if (!FULL && ra >= I_CNT) ra = I_CNT - 1;
    const float* hrow = h + (size_t)(U_CNT + ra) * DIM + col0;

    const int k0 = 2 * half;
    const int k1 = 1 + 2 * half;
#pragma unroll
    for (int j = 0; j < 4; ++j) {
        v2f a, b;
        a[0] = hrow[4 * j + k0];
        a[1] = hrow[4 * j + k1];
        // B (4x16) mirrors A with N in place of M:
        // lane -> N = n ; VGPR0 -> K = 2*half, VGPR1 -> K+1.
        b[0] = (n == 4 * j + k0) ? 0.75f : 0.0f;
        b[1] = (n == 4 * j + k1) ? 0.75f : 0.0f;
        acc = __builtin_amdgcn_wmma_f32_16x16x4_f32(
            /*neg_a=*/false, a, /*neg_b=*/false, b,
            /*c_mod=*/(short)0, acc, /*reuse_a=*/false, /*reuse_b=*/false);
    }

    float* obase = i_final + (size_t)mbase * DIM + col0 + n;
#pragma unroll
    for (int v = 0; v < 8; ++v) {
        if (FULL || (mbase + v) < I_CNT)
            obase[(size_t)v * DIM] = acc[v];
    }
}

__global__ void lgcn_finalize_items_wmma(const float* __restrict__ item_emb,
                                         const float* __restrict__ h,
                                         float*       __restrict__ i_final) {
    const int gid  = blockIdx.x * blockDim.x + threadIdx.x;
    const int tile = gid >> 5;          // uniform per wave
    const int lane = gid & 31;
    const int ntiles = ((I_CNT + 15) >> 4) * (DIM >> 4);
    if (tile >= ntiles) return;

    const int row0 = (tile >> 3) << 4;  // DIM/16 == 8 column tiles
    const int col0 = (tile & 7) << 4;

    if (row0 + 16 <= I_CNT)             // wave-uniform: 12496 of 12504 tiles
        lgcn_item_tile<true>(item_emb, h, i_final, row0, col0, lane);
    else                                // single tail row-tile per column tile
        lgcn_item_tile<false>(item_emb, h, i_final, row0, col0, lane);
}

// ----------------------------------------------------------- batch gathers ----
// u_g / pos_i_g / neg_i_g computed directly from ego + h (same blend).
__global__ void lgcn_gather(const float* __restrict__ user_emb,
                            const float* __restrict__ item_emb,
                            const float* __restrict__ h,
                            const int* __restrict__ users,
                            const int* __restrict__ pos_items,
                            const int* __restrict__ neg_items,
                            float* __restrict__ out) {
    const int gid  = blockIdx.x * blockDim.x + threadIdx.x;  // 3*BATCH waves
    const int lane = gid & 31;
    const int t    = gid >> 5;
    if (t >= 3 * BATCH) return;
    const int b   = t & (BATCH - 1);
    const int sec = t >> 12;            // BATCH == 4096 == 2^12

    const float* ego;
    const float* hr;
    float* o;
    if (sec == 0) {
        int u = users[b];
        ego = user_emb + (size_t)u * DIM;
        hr  = h + (size_t)u * DIM;
        o   = out + (size_t)b * DIM;
    } else if (sec == 1) {
        int p = pos_items[b];
        ego = item_emb + (size_t)p * DIM;
        hr  = h + (size_t)(U_CNT + p) * DIM;
        o   = out + (size_t)(BATCH + b) * DIM;
    } else {
        int q = neg_items[b];
        ego = item_emb + (size_t)q * DIM;
        hr  = h + (size_t)(U_CNT + q) * DIM;
        o   = out + (size_t)(2 * BATCH + b) * DIM;
    }

    const float4 e4 = *(const float4*)(ego + lane * 4);
    const float4 h4 = *(const float4*)(hr + lane * 4);
    float4 r4;
    r4.x = 0.25f * e4.x + 0.75f * h4.x;
    r4.y = 0.25f * e4.y + 0.75f * h4.y;
    r4.z = 0.25f * e4.z + 0.75f * h4.z;
    r4.w = 0.25f * e4.w + 0.75f * h4.w;
    *(float4*)(o + lane * 4) = r4;
}

// ------------------------------------------------------------------ launch ----
extern "C" void kernel_launch(void* const* d_in, const int* in_sizes, int n_in,
                              void* d_out, int out_size, void* d_ws, size_t ws_size,
                              hipStream_t stream) {
    const float* user_emb = (const float*)d_in[0];
    const float* item_emb = (const float*)d_in[1];
    const int*   adj_row  = (const int*)d_in[2];
    const int*   adj_col  = (const int*)d_in[3];
    const float* adj_vals = (const float*)d_in[4];
    const int*   users    = (const int*)d_in[5];
    const int*   pos_it   = (const int*)d_in[6];
    const int*   neg_it   = (const int*)d_in[7];

    const int nE = in_sizes[2];

    float* h = (float*)d_ws;                         // N_CNT * DIM floats (38.4 MB)
    float* out_u  = (float*)d_out;                   // [BATCH, DIM] x 3 sections
    float* out_if = out_u + (size_t)3 * BATCH * DIM; // [I_CNT, DIM]

    // 1) zero h
    {
        int n4 = N_CNT * DIM / 4;                    // 2,400,000
        lgcn_zero_f4<<<(n4 + 255) / 256, 256, 0, stream>>>((float4*)h, n4);
    }
    // 2) SpMM scatter-add (one wave per EDGES_PER_WAVE edges)
    {
        long long waves   = ((long long)nE + EDGES_PER_WAVE - 1) / EDGES_PER_WAVE;
        long long threads = waves * 32;
        int blocks = (int)((threads + 255) / 256);
        lgcn_spmm_edges<<<blocks, 256, 0, stream>>>(adj_row, adj_col, adj_vals,
                                                    user_emb, item_emb, h, nE);
    }
    // 3) i_final blend via WMMA (one wave per 16x16 tile)
    {
        int ntiles  = ((I_CNT + 15) >> 4) * (DIM >> 4);  // 1563 * 8 = 12504
        int threads = ntiles * 32;
        lgcn_finalize_items_wmma<<<(threads + 255) / 256, 256, 0, stream>>>(
            item_emb, h, out_if);
    }
    // 4) batch gathers (one wave per output row)
    {
        int threads = 3 * BATCH * 32;                // 393,216
        lgcn_gather<<<(threads + 255) / 256, 256, 0, stream>>>(
            user_emb, item_emb, h, users, pos_it, neg_it, out_u);
    }
}